// ECC_51900384805424
// MI455X (gfx1250) — compile-verified
//
#include <hip/hip_runtime.h>

// Fused NNConv GNN for MI455X (gfx1250).
// One workgroup (1024 threads = 32 wave32) per graph; h and agg stay in LDS
// (320KB/WGP) across all 3 layers; edge scatter uses ds_add_f32 LDS atomics;
// layer GEMMs use V_WMMA_F32_16X16X4_F32 (full f32, matches reference dtype).

#define DFEAT    32
#define RLAYERS  3
#define NCLS     10
#define NPG      512
#define EPG      8192
#define HSTRIDE  33      // padded feature stride -> conflict-free LDS banks
#define WSTRIDE  33
#define NTHREADS 1024

typedef __attribute__((ext_vector_type(2))) float v2f;
typedef __attribute__((ext_vector_type(8))) float v8f;

__device__ __forceinline__ float lrelu(float v) { return v > 0.0f ? v : 0.01f * v; }

__global__ __launch_bounds__(NTHREADS) void ecc_fused(
    const float* __restrict__ x,
    const int*   __restrict__ edge_src,
    const int*   __restrict__ edge_dst,
    const float* __restrict__ We,
    const float* __restrict__ be,
    const float* __restrict__ bconv,
    const float* __restrict__ Wc1,
    const float* __restrict__ bc1,
    const float* __restrict__ Wc2,
    const float* __restrict__ bc2,
    float* __restrict__ out)
{
    __shared__ float h_buf[NPG * HSTRIDE];     // 67584 B
    __shared__ float agg_buf[NPG * HSTRIDE];   // 67584 B
    __shared__ float Wj[DFEAT * WSTRIDE];      //  4224 B
    __shared__ float pooled[DFEAT];
    __shared__ float z1[DFEAT];

    const int g    = blockIdx.x;
    const int t    = threadIdx.x;
    const int lane = t & 31;
    const int wave = t >> 5;

    // ---- load this graph's node features, fully coalesced float4 ----
    {
        const float4* xg = (const float4*)(x + (size_t)g * NPG * DFEAT);
#pragma unroll
        for (int k = 0; k < (NPG * DFEAT / 4) / NTHREADS; ++k) {   // 4 iters
            int    i = t + k * NTHREADS;        // float4 index (4096 total)
            float4 v = xg[i];
            int node = i >> 3;                  // 8 float4 per node
            int f    = (i & 7) << 2;
            float* p = &h_buf[node * HSTRIDE + f];
            p[0] = v.x; p[1] = v.y; p[2] = v.z; p[3] = v.w;
        }
    }
    if (t < DFEAT) pooled[t] = 0.0f;

    const int ebase = g * EPG;
    const int nbase = g * NPG;

    for (int j = 0; j < RLAYERS; ++j) {
        // zero aggregation buffer
        for (int i = t; i < NPG * HSTRIDE; i += NTHREADS) agg_buf[i] = 0.0f;
        // shared edge-weight matrix: W_j = lrelu(0.01*We_j + be_j)  (1024 elems)
        {
            int kk = t >> 5, nn = t & 31;
            const float* Wsrc = We + j * DFEAT * DFEAT;
            const float* bsrc = be + j * DFEAT * DFEAT;
            Wj[kk * WSTRIDE + nn] = lrelu(0.01f * Wsrc[t] + bsrc[t]);
        }
        __syncthreads();

        // ---- edge scatter: agg[dst] += h[src] ----
        // Each wave owns 256 consecutive edges; lanes load 32 edges at once
        // (coalesced), then one edge per step with lanes covering features.
        for (int blk = 0; blk < EPG / NTHREADS; ++blk) {           // 8 iters
            int e = ebase + wave * (EPG / 32) + blk * 32 + lane;
            int s = edge_src[e] - nbase;
            int d = edge_dst[e] - nbase;
#pragma unroll
            for (int q = 0; q < 32; ++q) {
                int sq = __builtin_amdgcn_readlane(s, q);          // scalar bcast
                int dq = __builtin_amdgcn_readlane(d, q);
                float v = h_buf[sq * HSTRIDE + lane];              // conflict-free
                atomicAdd(&agg_buf[dq * HSTRIDE + lane], v);       // ds_add_f32
            }
        }
        __syncthreads();

        // ---- h = lrelu(agg @ Wj + bconv_j) via V_WMMA_F32_16X16X4_F32 ----
        // wave w computes output rows [16w, 16w+16), both 16-col halves.
        {
            int mt    = wave;
            int row   = mt * 16 + (lane & 15);
            int khalf = (lane >> 4) * 2;                // lanes16-31: K+2/K+3
#pragma unroll
            for (int nt = 0; nt < 2; ++nt) {
                int n0  = nt * 16;
                int col = n0 + (lane & 15);
                v8f c = {};
#pragma unroll
                for (int k0 = 0; k0 < DFEAT; k0 += 4) {
                    v2f a, b;
                    a.x = agg_buf[row * HSTRIDE + k0 + khalf];
                    a.y = agg_buf[row * HSTRIDE + k0 + khalf + 1];
                    b.x = Wj[(k0 + khalf) * WSTRIDE + col];
                    b.y = Wj[(k0 + khalf + 1) * WSTRIDE + col];
                    c = __builtin_amdgcn_wmma_f32_16x16x4_f32(
                        false, a, false, b, (short)0, c, false, false);
                }
                float bcv  = bconv[j * DFEAT + col];
                int   rbase = mt * 16 + (lane >> 4) * 8;   // D: M = v + 8*hi
#pragma unroll
                for (int v = 0; v < 8; ++v)
                    h_buf[(rbase + v) * HSTRIDE + col] = lrelu(c[v] + bcv);
            }
        }
        __syncthreads();
    }

    // ---- per-graph sum pooling: wave w sums 16 nodes, lane = feature ----
    {
        float acc = 0.0f;
        int n0 = wave * (NPG / 32);
#pragma unroll
        for (int n = 0; n < NPG / 32; ++n)
            acc += h_buf[(n0 + n) * HSTRIDE + lane];
        atomicAdd(&pooled[lane], acc);
    }
    __syncthreads();

    // ---- classifier MLP: [1,32] -> lrelu(@Wc1+bc1) -> @Wc2+bc2 -> [10] ----
    if (t < DFEAT) {
        float s = bc1[t];
        for (int k = 0; k < DFEAT; ++k) s += pooled[k] * Wc1[k * DFEAT + t];
        z1[t] = lrelu(s);
    }
    __syncthreads();
    if (t < NCLS) {
        float s = bc2[t];
        for (int f = 0; f < DFEAT; ++f) s += z1[f] * Wc2[f * NCLS + t];
        out[g * NCLS + t] = s;
    }
}

extern "C" void kernel_launch(void* const* d_in, const int* in_sizes, int n_in,
                              void* d_out, int out_size, void* d_ws, size_t ws_size,
                              hipStream_t stream) {
    const float* x     = (const float*)d_in[0];
    const int*   esrc  = (const int*)d_in[1];
    const int*   edst  = (const int*)d_in[2];
    // d_in[3] = node_graph_id (unused: batched layout is implicit)
    const float* We    = (const float*)d_in[4];
    const float* be    = (const float*)d_in[5];
    const float* bconv = (const float*)d_in[6];
    const float* Wc1   = (const float*)d_in[7];
    const float* bc1   = (const float*)d_in[8];
    const float* Wc2   = (const float*)d_in[9];
    const float* bc2   = (const float*)d_in[10];
    float* out = (float*)d_out;

    ecc_fused<<<dim3(32), dim3(NTHREADS), 0, stream>>>(
        x, esrc, edst, We, be, bconv, Wc1, bc1, Wc2, bc2, out);
}